// Qwen3_5Attention_3796751089958
// MI455X (gfx1250) — compile-verified
//
#include <hip/hip_runtime.h>
#include <math.h>

typedef __attribute__((ext_vector_type(2))) float v2f;
typedef __attribute__((ext_vector_type(4))) float v4f;
typedef __attribute__((ext_vector_type(8))) float v8f;
typedef __attribute__((ext_vector_type(4))) unsigned int v4u;
typedef __attribute__((ext_vector_type(4))) int v4i;
typedef __attribute__((ext_vector_type(8))) int v8i;

#define H_    2048
#define NH_   16
#define NKV_  4
#define HD_   128
#define G_    4
#define B_    32
#define S_    4096
#define QKVN  ((NH_ + 2 * NKV_) * HD_)   /* 3072 */
#define SCALE_ 0.08838834764831845f      /* 128^-0.5 */
#define KVROW 1024                       /* floats between seq positions */
#define NW    4                          /* waves per attention block */

#if defined(__gfx1250__) && __has_builtin(__builtin_amdgcn_tensor_load_to_lds)
#define HAVE_TDM 1
#else
#define HAVE_TDM 0
#endif

__device__ __forceinline__ v8f wmma4(v2f a, v2f b, v8f c) {
  // D(16x16,f32) = A(16x4,f32) x B(4x16,f32) + C
  return __builtin_amdgcn_wmma_f32_16x16x4_f32(false, a, false, b, (short)0, c,
                                               false, false);
}

// One wave computes a 32x16 tile of Out(32xN) = A(32x2048) * W(2048xN) (+bias)
__global__ __launch_bounds__(32) void gemm32(const float* __restrict__ Amat,
                                             const float* __restrict__ Wmat,
                                             const float* __restrict__ bias,
                                             float* __restrict__ Out, int N) {
  const int l  = threadIdx.x;
  const int lr = l & 15;
  const int kh = l >> 4;
  const int n  = blockIdx.x * 16 + lr;

  const v8f zero = {0.f, 0.f, 0.f, 0.f, 0.f, 0.f, 0.f, 0.f};
  v8f c0 = zero, c1 = zero;

  const float* a0p = Amat + (size_t)lr * 2048;
  const float* a1p = Amat + (size_t)(lr + 16) * 2048;

#pragma unroll 4
  for (int k4 = 0; k4 < 512; ++k4) {
    const int k = k4 * 4 + 2 * kh;
    v2f a0 = *(const v2f*)(a0p + k);
    v2f a1 = *(const v2f*)(a1p + k);
    v2f b;
    b.x = Wmat[(size_t)k * N + n];
    b.y = Wmat[(size_t)(k + 1) * N + n];
    c0 = wmma4(a0, b, c0);
    c1 = wmma4(a1, b, c1);
  }

  const float bq = bias ? bias[n] : 0.0f;
#pragma unroll
  for (int v = 0; v < 8; ++v) {
    const int m0 = v + 8 * kh;  // C/D layout: lane l, vgpr v -> M = v + 8*(l/16)
    Out[(size_t)m0 * N + n]        = c0[v] + bq;
    Out[(size_t)(m0 + 16) * N + n] = c1[v] + bq;
  }
}

#if HAVE_TDM
// TDM: DMA one 16x128 f32 tile (row stride KVROW floats) from global into LDS.
__device__ __forceinline__ void tdm_load_16x128(const float* gsrc,
                                                float* lds_dst) {
  const unsigned long long ga = (unsigned long long)(size_t)gsrc;
  const unsigned int laddr = (unsigned int)(size_t)lds_dst;  // addr[31:0] = LDS byte offset
  v4u g0;
  g0.x = 1u;                                   // count=1 (user descriptor)
  g0.y = laddr;                                // lds_addr  (bits 63:32)
  g0.z = (unsigned int)ga;                     // global_addr[31:0]
  g0.w = ((unsigned int)(ga >> 32) & 0x01FFFFFFu) | (2u << 30);  // addr[56:32] | type=2
  v8i g1;
  g1[0] = (int)(2u << 16);    // workgroup_mask=0, data_size=2 (4 bytes)
  g1[1] = (int)(128u << 16);  // tensor_dim0 = 128  (bits 79:48, low half)
  g1[2] = (int)(16u << 16);   // tensor_dim0 hi=0 | tensor_dim1 = 16
  g1[3] = (int)(128u << 16);  // tensor_dim1 hi=0 | tile_dim0 = 128
  g1[4] = 16;                 // tile_dim1 = 16, tile_dim2 = 0
  g1[5] = KVROW;              // tensor_dim0_stride = 1024 elements (low 32)
  g1[6] = 0;                  // stride0 hi | tensor_dim1_stride lo
  g1[7] = 0;
  const v4i z4 = {0, 0, 0, 0};                   // groups 2/3 unused (2-D tensor)
  const v8i z8 = {0, 0, 0, 0, 0, 0, 0, 0};
  // clang-23 / therock-10.0 arity: (g0, g1, g2, g3, g4, cpol)
  __builtin_amdgcn_tensor_load_to_lds(g0, g1, z4, z4, z8, 0);
}
#endif

// One block per (batch, kv_head); NW waves flash-attention over the KV cache,
// K/V tiles staged in LDS by the Tensor Data Mover (double buffered).
__global__ __launch_bounds__(32 * NW) void attn_kernel(
    const float* __restrict__ qkv, const float* __restrict__ kv,
    const int* __restrict__ slot_mapping, const int* __restrict__ seq_lens,
    float* __restrict__ attn_out) {
  const int b   = blockIdx.x >> 2;
  const int kvh = blockIdx.x & 3;
  const int tid = threadIdx.x;
  const int w   = tid >> 5;
  const int l   = tid & 31;
  const int lr  = l & 15;
  const int kh  = l >> 4;

  __shared__ float sK[NW][2][16][HD_];   // 64 KB: double-buffered K tiles
  __shared__ float sV[NW][2][16][HD_];   // 64 KB: double-buffered V tiles
  __shared__ float sQ[G_][HD_];          // pre-scaled Q
  __shared__ float pP[NW][G_][16];       // per-wave P tile (query-major)
  __shared__ float red_m[NW][G_];
  __shared__ float red_l[NW][G_];
  __shared__ float oPart[NW][G_][HD_];   // per-wave unnormalized partial O

  const int seq  = seq_lens[b];
  const int slot = slot_mapping[b];

  const float* qbase = qkv + (size_t)b * QKVN + (size_t)(kvh * G_) * HD_;
  const float* knew  = qkv + (size_t)b * QKVN + NH_ * HD_ + kvh * HD_;
  const float* vnew  = knew + NKV_ * HD_;
  const float* kbase = kv + (size_t)b * ((size_t)S_ * KVROW) + (size_t)kvh * HD_;
  const float* vbase = kbase + NKV_ * HD_;  // c=1 plane

  for (int i = tid; i < G_ * HD_; i += 32 * NW)
    ((float*)sQ)[i] = qbase[i] * SCALE_;
  __syncthreads();

  const v8f zero = {0.f, 0.f, 0.f, 0.f, 0.f, 0.f, 0.f, 0.f};
  v8f acc[8];
#pragma unroll
  for (int c = 0; c < 8; ++c) acc[c] = zero;
  float mM[G_] = {-__builtin_inff(), -__builtin_inff(), -__builtin_inff(),
                  -__builtin_inff()};
  float lS[G_] = {0.f, 0.f, 0.f, 0.f};

  const int nt = (seq + 15) >> 4;

#if HAVE_TDM
  if (w < nt) {  // prologue: prime buffer 0
    tdm_load_16x128(kbase + (size_t)(w * 16) * KVROW, &sK[w][0][0][0]);
    tdm_load_16x128(vbase + (size_t)(w * 16) * KVROW, &sV[w][0][0][0]);
  }
#endif

  int bi = 0;
  for (int tile = w; tile < nt; tile += NW) {
    const int t16 = tile * 16;

#if HAVE_TDM
    const int tnext = tile + NW;
    if (tnext < nt) {  // issue next tile into the other buffer, then drain cur
      tdm_load_16x128(kbase + (size_t)(tnext * 16) * KVROW, &sK[w][bi ^ 1][0][0]);
      tdm_load_16x128(vbase + (size_t)(tnext * 16) * KVROW, &sV[w][bi ^ 1][0][0]);
      __builtin_amdgcn_s_wait_tensorcnt((short)2);
    } else {
      __builtin_amdgcn_s_wait_tensorcnt((short)0);
    }
    __builtin_amdgcn_fence(__ATOMIC_ACQUIRE, "workgroup");
#else
    // fallback: coalesced manual staging, one full 512B row per instruction
#pragma unroll
    for (int j = 0; j < 16; ++j) {
      *(v4f*)&sK[w][bi][j][l * 4] =
          *(const v4f*)(kbase + (size_t)(t16 + j) * KVROW + l * 4);
      *(v4f*)&sV[w][bi][j][l * 4] =
          *(const v4f*)(vbase + (size_t)(t16 + j) * KVROW + l * 4);
    }
#endif
    // patch the current-token row with the freshly computed K/V (cache is input-only)
    if (slot >= t16 && slot < t16 + 16) {
      const int r = slot - t16;
      *(v4f*)&sK[w][bi][r][l * 4] = *(const v4f*)(knew + l * 4);
      *(v4f*)&sV[w][bi][r][l * 4] = *(const v4f*)(vnew + l * 4);
    }
    __builtin_amdgcn_fence(__ATOMIC_ACQ_REL, "wavefront");

    // S(16q x 16k) = Q(16x128) * K^T(128x16), K-dim in chunks of 4
    v8f sc = zero;
#pragma unroll
    for (int kk = 0; kk < 32; ++kk) {
      const int hd0 = kk * 4 + 2 * kh;
      v2f aq = *(const v2f*)&sQ[l & 3][hd0];  // A row q=l%16 (rows 4..15 dup, unused)
      v2f bk = *(const v2f*)&sK[w][bi][lr][hd0];
      sc = wmma4(aq, bk, sc);
    }

    const bool kvalid = (t16 + lr < seq);
    float esc[G_];
#pragma unroll
    for (int v = 0; v < G_; ++v) {  // query v lives in vgpr v, lanes 0..15
      float sv = kvalid ? sc[v] : -1e30f;
      float t = sv;
      t = fmaxf(t, __shfl_xor(t, 1, 32));
      t = fmaxf(t, __shfl_xor(t, 2, 32));
      t = fmaxf(t, __shfl_xor(t, 4, 32));
      t = fmaxf(t, __shfl_xor(t, 8, 32));
      const float mnew = fmaxf(mM[v], t);
      const float e = __expf(mM[v] - mnew);
      const float p = __expf(sv - mnew);
      float ps = p;
      ps += __shfl_xor(ps, 1, 32);
      ps += __shfl_xor(ps, 2, 32);
      ps += __shfl_xor(ps, 4, 32);
      ps += __shfl_xor(ps, 8, 32);
      lS[v] = lS[v] * e + ps;
      mM[v] = mnew;
      esc[v] = e;
      if (l < 16) pP[w][v][lr] = p;
    }
#pragma unroll
    for (int c = 0; c < 8; ++c) {
      acc[c][0] *= esc[0];
      acc[c][1] *= esc[1];
      acc[c][2] *= esc[2];
      acc[c][3] *= esc[3];
    }
    __builtin_amdgcn_fence(__ATOMIC_ACQ_REL, "wavefront");

    // O(16q x 128d) += P(16x16) * V(16x128); K-dim = keys in chunks of 4
#pragma unroll
    for (int kk = 0; kk < 4; ++kk) {
      const int key0 = kk * 4 + 2 * kh;
      v2f ap = *(const v2f*)&pP[w][l & 3][key0];
#pragma unroll
      for (int c = 0; c < 8; ++c) {
        const int d = lr + 16 * c;
        v2f bv;
        bv.x = sV[w][bi][key0][d];
        bv.y = sV[w][bi][key0 + 1][d];
        acc[c] = wmma4(ap, bv, acc[c]);
      }
    }
    bi ^= 1;
  }

  if (l < 16) {
#pragma unroll
    for (int c = 0; c < 8; ++c)
#pragma unroll
      for (int v = 0; v < G_; ++v) oPart[w][v][lr + 16 * c] = acc[c][v];
  }
  if (l == 0) {
#pragma unroll
    for (int v = 0; v < G_; ++v) {
      red_m[w][v] = mM[v];
      red_l[w][v] = lS[v];
    }
  }
  __syncthreads();

  // merge the NW per-wave softmax partials
  for (int e = tid; e < G_ * HD_; e += 32 * NW) {
    const int q = e >> 7;
    const int d = e & 127;
    float M = -__builtin_inff();
#pragma unroll
    for (int ww = 0; ww < NW; ++ww) M = fmaxf(M, red_m[ww][q]);
    float num = 0.f, den = 0.f;
#pragma unroll
    for (int ww = 0; ww < NW; ++ww) {
      const float f = __expf(red_m[ww][q] - M);
      num += f * oPart[ww][q][d];
      den += f * red_l[ww][q];
    }
    attn_out[(size_t)b * (NH_ * HD_) + (size_t)(kvh * G_ + q) * HD_ + d] =
        num / den;
  }
}

extern "C" void kernel_launch(void* const* d_in, const int* in_sizes, int n_in,
                              void* d_out, int out_size, void* d_ws,
                              size_t ws_size, hipStream_t stream) {
  const float* hs   = (const float*)d_in[0];
  // d_in[1] = positions (unused; equals seq_lens-1)
  const float* kvc  = (const float*)d_in[2];
  const int* slotm  = (const int*)d_in[3];
  const int* seqs   = (const int*)d_in[4];
  const float* Wqkv = (const float*)d_in[5];
  const float* bqkv = (const float*)d_in[6];
  const float* Wo   = (const float*)d_in[7];
  float* out = (float*)d_out;

  float* qkv_ws  = (float*)d_ws;            // 32*3072 floats
  float* attn_ws = qkv_ws + B_ * QKVN;      // 32*2048 floats

  gemm32<<<dim3(QKVN / 16), dim3(32), 0, stream>>>(hs, Wqkv, bqkv, qkv_ws, QKVN);
  attn_kernel<<<dim3(B_ * NKV_), dim3(32 * NW), 0, stream>>>(qkv_ws, kvc, slotm,
                                                             seqs, attn_ws);
  gemm32<<<dim3(H_ / 16), dim3(32), 0, stream>>>(attn_ws, Wo, nullptr, out, H_);
}